// MultiHeadAttention_35055523070637
// MI455X (gfx1250) — compile-verified
//
#include <hip/hip_runtime.h>
#include <cstdint>

typedef float v2f __attribute__((ext_vector_type(2)));
typedef float v8f __attribute__((ext_vector_type(8)));
typedef float f4v __attribute__((ext_vector_type(4)));

#define D_MODEL 1024
#define SEQ     2048
#define NHEAD   16
#define DK      64
#define MTOK    4096      // B * S
#define LDSS    36        // padded LDS row stride (floats): 16B-aligned, bank-conflict free

__device__ __forceinline__ v8f wmma_f32(v2f a, v2f b, v8f c) {
  // V_WMMA_F32_16X16X4_F32 : D = A(16x4 f32) * B(4x16 f32) + C(16x16 f32)
  return __builtin_amdgcn_wmma_f32_16x16x4_f32(false, a, false, b, (short)0, c,
                                               false, false);
}

// --- CDNA5 async global->LDS helpers (ASYNCcnt-tracked DMA, no VGPR bounce) --
__device__ __forceinline__ unsigned lds_off(const void* p) {
  // generic LDS pointer: addr[31:0] is the LDS byte offset (aperture in hi bits)
  return (unsigned)(uintptr_t)p;
}
__device__ __forceinline__ void async_ld128(unsigned lds_byte, const float* g) {
  asm volatile("global_load_async_to_lds_b128 %0, %1, off"
               :: "v"(lds_byte), "v"((unsigned long long)(uintptr_t)g)
               : "memory");
}
__device__ __forceinline__ void async_ld32(unsigned lds_byte, const float* g) {
  asm volatile("global_load_async_to_lds_b32 %0, %1, off"
               :: "v"(lds_byte), "v"((unsigned long long)(uintptr_t)g)
               : "memory");
}
template <int N>
__device__ __forceinline__ void wait_async() {
  asm volatile("s_wait_asynccnt %0" :: "i"(N) : "memory");
}

// ---------------------------------------------------------------------------
// Double-buffered GEMM core:  Out(128 x 32) tile of  A[M,K] @ W[N,K]^T
// 8 waves: wave w owns rows m0+w*16..+15, each wave computes 2 N-subtiles.
// Chunk c+1 staged to LDS via async DMA while chunk c is consumed by WMMA.
// 5 async b128 per thread per chunk (4 for A-tile 128x32, 1 for B-tile 32x32).
// ---------------------------------------------------------------------------
__device__ __forceinline__ void gemm_issue(const float* __restrict__ Arows,
                                           const float* __restrict__ Brows,
                                           float* As, float* Bs, int k0) {
  const int tid = threadIdx.x;
  const int r8 = tid >> 3, seg = (tid & 7) * 4;
#pragma unroll
  for (int p = 0; p < 4; ++p) {
    const int r = p * 32 + r8;
    async_ld128(lds_off(&As[r * LDSS + seg]),
                Arows + (size_t)r * 1024 + k0 + seg);
  }
  async_ld128(lds_off(&Bs[r8 * LDSS + seg]),
              Brows + (size_t)r8 * 1024 + k0 + seg);
}

__device__ __forceinline__ void gemm_compute(const float* As, const float* Bs,
                                             v8f acc[2], int w, int lh, int ll) {
#pragma unroll
  for (int kk = 0; kk < 32; kk += 4) {
    v2f a = *(const v2f*)&As[(w * 16 + ll) * LDSS + kk + 2 * lh];
#pragma unroll
    for (int nt = 0; nt < 2; ++nt) {
      v2f b = *(const v2f*)&Bs[((nt << 4) + ll) * LDSS + kk + 2 * lh];
      acc[nt] = wmma_f32(a, b, acc[nt]);
    }
  }
}

__device__ __forceinline__ void gemm_mainloop(const float* __restrict__ Arows,
                                              const float* __restrict__ Brows,
                                              float (*As)[128 * LDSS],
                                              float (*Bs)[32 * LDSS],
                                              v8f acc[2]) {
  const int lane = threadIdx.x & 31, w = threadIdx.x >> 5;
  const int lh = lane >> 4, ll = lane & 15;
  gemm_issue(Arows, Brows, As[0], Bs[0], 0);
  for (int c = 0; c < 32; ++c) {  // 1024 / 32 K-chunks
    const int cur = c & 1;
    if (c + 1 < 32) {
      gemm_issue(Arows, Brows, As[cur ^ 1], Bs[cur ^ 1], (c + 1) * 32);
      wait_async<5>();   // chunk c landed; only chunk c+1's 5 DMAs in flight
    } else {
      wait_async<0>();
    }
    __syncthreads();     // all waves' chunk-c tiles visible in LDS
    gemm_compute(As[cur], Bs[cur], acc, w, lh, ll);
    __syncthreads();     // all waves done reading before buffer reuse
  }
}

// ---------------------------------------------------------------------------
// Kernel 1: Q/K/V projections.  grid=(32 n-blocks, 32 m-blocks, 3 matrices)
// Q gets the 1/sqrt(DK)=0.125 scale folded in.  Output layout [B,H,S,DK].
// ---------------------------------------------------------------------------
__global__ __launch_bounds__(256) void qkv_proj_kernel(
    const float* __restrict__ x, const float* __restrict__ wq,
    const float* __restrict__ wk, const float* __restrict__ wv,
    float* __restrict__ Qo, float* __restrict__ Ko, float* __restrict__ Vo) {
  __shared__ float As[2][128 * LDSS];
  __shared__ float Bs[2][32 * LDSS];
  const int which = blockIdx.z;
  const float* W = (which == 0) ? wq : (which == 1) ? wk : wv;
  float* Out     = (which == 0) ? Qo : (which == 1) ? Ko : Vo;
  const float scale = (which == 0) ? 0.125f : 1.0f;
  const int n0 = blockIdx.x * 32, m0 = blockIdx.y * 128;
  v8f acc[2] = {};
  gemm_mainloop(x + (size_t)m0 * D_MODEL, W + (size_t)n0 * D_MODEL, As, Bs, acc);
  const int lane = threadIdx.x & 31, w = threadIdx.x >> 5;
  const int lh = lane >> 4, ll = lane & 15;
#pragma unroll
  for (int nt = 0; nt < 2; ++nt)
#pragma unroll
    for (int i = 0; i < 8; ++i) {
      const int m = m0 + w * 16 + i + 8 * lh;
      const int n = n0 + nt * 16 + ll;
      const int b = m >> 11, s = m & (SEQ - 1);
      const int h = n >> 6, d = n & (DK - 1);
      Out[(((size_t)(b * NHEAD + h)) * SEQ + s) * DK + d] = acc[nt][i] * scale;
    }
}

// ---------------------------------------------------------------------------
// Kernel 2: scores = Qs @ K^T, key-padding mask, softmax, write attn once.
// Block = 8 waves, 16 query rows, full 2048 keys register-resident
// (16 tiles x v8f = 128 VGPRs per wave).  grid=(128, 16, 2)
// ---------------------------------------------------------------------------
__global__ __launch_bounds__(256) void attn_softmax_kernel(
    const float* __restrict__ Qm, const float* __restrict__ Km,
    const unsigned char* __restrict__ mask, float* __restrict__ attn) {
  __shared__ float Qs[16 * 68];
  __shared__ float red[16 * 8];
  const int tid = threadIdx.x, lane = tid & 31, w = tid >> 5;
  const int lh = lane >> 4, ll = lane & 15;
  const int s0 = blockIdx.x * 16;
  const int h = blockIdx.y, b = blockIdx.z;
  const float* Qg = Qm + ((size_t)(b * NHEAD + h) * SEQ + s0) * DK;
  const float* Kg = Km + ((size_t)(b * NHEAD + h) * SEQ) * DK;
  {  // stage 16x64 Q tile (coalesced float4, padded stride 68)
    const int r = tid >> 4, seg = (tid & 15) * 4;
    *(f4v*)&Qs[r * 68 + seg] = *(const f4v*)&Qg[r * DK + seg];
  }
  __syncthreads();
  const unsigned char* mrow = mask + (size_t)b * SEQ;

  v8f c[16] = {};
#pragma unroll
  for (int j = 0; j < 16; ++j) {
    const int t0 = (j * 8 + w) * 16;
    const float* kp = Kg + (size_t)(t0 + ll) * DK + 2 * lh;
    if (j < 15) {  // prefetch next K tile row (L2-resident; hides first touch)
      __builtin_prefetch(Kg + (size_t)(t0 + 128 + ll) * DK + 2 * lh, 0, 1);
    }
#pragma unroll
    for (int k0 = 0; k0 < DK; k0 += 4) {
      v2f a  = *(const v2f*)&Qs[ll * 68 + k0 + 2 * lh];
      v2f bv = *(const v2f*)&kp[k0];
      c[j] = wmma_f32(a, bv, c[j]);
    }
    if (mrow[t0 + ll]) {  // key-padding mask (True => -1e9 before softmax)
#pragma unroll
      for (int i = 0; i < 8; ++i) c[j][i] = -1e9f;
    }
  }
  // ---- row max: in-register across tiles, 16-lane butterfly, cross-wave LDS
  float rstat[8];
#pragma unroll
  for (int i = 0; i < 8; ++i) {
    float v = c[0][i];
#pragma unroll
    for (int j = 1; j < 16; ++j) v = fmaxf(v, c[j][i]);
    v = fmaxf(v, __shfl_xor(v, 1, 16));
    v = fmaxf(v, __shfl_xor(v, 2, 16));
    v = fmaxf(v, __shfl_xor(v, 4, 16));
    v = fmaxf(v, __shfl_xor(v, 8, 16));
    rstat[i] = v;
  }
  if (ll == 0) {
#pragma unroll
    for (int i = 0; i < 8; ++i) red[(i + 8 * lh) * 8 + w] = rstat[i];
  }
  __syncthreads();
#pragma unroll
  for (int i = 0; i < 8; ++i) {
    const int m = i + 8 * lh;
    float v = red[m * 8];
#pragma unroll
    for (int ww = 1; ww < 8; ++ww) v = fmaxf(v, red[m * 8 + ww]);
    rstat[i] = v;
  }
  __syncthreads();
  // ---- exp + row sum
  float rsum[8] = {0.f, 0.f, 0.f, 0.f, 0.f, 0.f, 0.f, 0.f};
#pragma unroll
  for (int j = 0; j < 16; ++j)
#pragma unroll
    for (int i = 0; i < 8; ++i) {
      const float e = __expf(c[j][i] - rstat[i]);
      c[j][i] = e;
      rsum[i] += e;
    }
#pragma unroll
  for (int i = 0; i < 8; ++i) {
    float v = rsum[i];
    v += __shfl_xor(v, 1, 16);
    v += __shfl_xor(v, 2, 16);
    v += __shfl_xor(v, 4, 16);
    v += __shfl_xor(v, 8, 16);
    rsum[i] = v;
  }
  if (ll == 0) {
#pragma unroll
    for (int i = 0; i < 8; ++i) red[(i + 8 * lh) * 8 + w] = rsum[i];
  }
  __syncthreads();
#pragma unroll
  for (int i = 0; i < 8; ++i) {
    const int m = i + 8 * lh;
    float v = red[m * 8];
#pragma unroll
    for (int ww = 1; ww < 8; ++ww) v += red[m * 8 + ww];
    rstat[i] = 1.0f / v;  // inverse row sum
  }
  // ---- normalized attn store (lanes 0-15 write 64B contiguous per op)
  float* arow = attn + ((size_t)(b * NHEAD + h) * SEQ + s0) * SEQ;
#pragma unroll
  for (int j = 0; j < 16; ++j) {
    const int t0 = (j * 8 + w) * 16;
#pragma unroll
    for (int i = 0; i < 8; ++i) {
      const int m = i + 8 * lh;
      arow[(size_t)m * SEQ + t0 + ll] = c[j][i] * rstat[i];
    }
  }
}

// ---------------------------------------------------------------------------
// Kernel 3: ctx = attn @ V.  Block tile 128(queries) x 64(Dv), K=2048 in
// double-buffered async-DMA chunks of 32.  The V tile is transposed by the
// per-lane LDS addressing of global_load_async_to_lds_b32 (scatter for free).
// 12 async ops per thread per chunk (4x b128 attn + 8x b32 V).  grid=(16,16,2)
// ---------------------------------------------------------------------------
__device__ __forceinline__ void av_issue(const float* __restrict__ Arow,
                                         const float* __restrict__ Vg,
                                         float* As, float* Vt, int k0) {
  const int tid = threadIdx.x;
  {
    const int r8 = tid >> 3, seg = (tid & 7) * 4;
#pragma unroll
    for (int p = 0; p < 4; ++p) {  // attn tile 128x32
      const int r = p * 32 + r8;
      async_ld128(lds_off(&As[r * LDSS + seg]),
                  Arow + (size_t)r * SEQ + k0 + seg);
    }
  }
#pragma unroll
  for (int q = 0; q < 8; ++q) {  // V tile 32x64 -> Vt[n][k] transposed scatter
    const int e = q * 256 + tid;
    const int r = e >> 6, cl = e & 63;
    async_ld32(lds_off(&Vt[cl * LDSS + r]), Vg + (size_t)(k0 + r) * DK + cl);
  }
}

__global__ __launch_bounds__(256) void attn_v_kernel(
    const float* __restrict__ attn, const float* __restrict__ Vm,
    float* __restrict__ ctx) {
  __shared__ float As[2][128 * LDSS];
  __shared__ float Vt[2][64 * LDSS];
  const int tid = threadIdx.x, lane = tid & 31, w = tid >> 5;
  const int lh = lane >> 4, ll = lane & 15;
  const int m0 = blockIdx.x * 128;
  const int h = blockIdx.y, b = blockIdx.z;
  const float* Arow = attn + ((size_t)(b * NHEAD + h) * SEQ + m0) * SEQ;
  const float* Vg   = Vm + ((size_t)(b * NHEAD + h) * SEQ) * DK;
  v8f acc[4] = {};
  av_issue(Arow, Vg, As[0], Vt[0], 0);
  for (int c = 0; c < SEQ / 32; ++c) {
    const int cur = c & 1;
    if (c + 1 < SEQ / 32) {
      av_issue(Arow, Vg, As[cur ^ 1], Vt[cur ^ 1], (c + 1) * 32);
      wait_async<12>();
    } else {
      wait_async<0>();
    }
    __syncthreads();
#pragma unroll
    for (int kk = 0; kk < 32; kk += 4) {
      v2f a = *(const v2f*)&As[cur][(w * 16 + ll) * LDSS + kk + 2 * lh];
#pragma unroll
      for (int nt = 0; nt < 4; ++nt) {
        v2f bv = *(const v2f*)&Vt[cur][(nt * 16 + ll) * LDSS + kk + 2 * lh];
        acc[nt] = wmma_f32(a, bv, acc[nt]);
      }
    }
    __syncthreads();
  }
#pragma unroll
  for (int nt = 0; nt < 4; ++nt)
#pragma unroll
    for (int i = 0; i < 8; ++i) {
      const int s = m0 + w * 16 + i + 8 * lh;
      const int n = nt * 16 + ll;
      ctx[((size_t)(b * SEQ + s)) * D_MODEL + h * DK + n] = acc[nt][i];
    }
}

// ---------------------------------------------------------------------------
// Kernel 4: pre = ctx @ w_fc^T + x (residual).  grid=(32, 32, 1)
// ---------------------------------------------------------------------------
__global__ __launch_bounds__(256) void fc_proj_kernel(
    const float* __restrict__ ctx, const float* __restrict__ wfc,
    const float* __restrict__ x, float* __restrict__ pre) {
  __shared__ float As[2][128 * LDSS];
  __shared__ float Bs[2][32 * LDSS];
  const int n0 = blockIdx.x * 32, m0 = blockIdx.y * 128;
  v8f acc[2] = {};
  gemm_mainloop(ctx + (size_t)m0 * D_MODEL, wfc + (size_t)n0 * D_MODEL, As, Bs,
                acc);
  const int lane = threadIdx.x & 31, w = threadIdx.x >> 5;
  const int lh = lane >> 4, ll = lane & 15;
#pragma unroll
  for (int nt = 0; nt < 2; ++nt)
#pragma unroll
    for (int i = 0; i < 8; ++i) {
      const int m = m0 + w * 16 + i + 8 * lh;
      const int n = n0 + nt * 16 + ll;
      pre[(size_t)m * D_MODEL + n] = acc[nt][i] + x[(size_t)m * D_MODEL + n];
    }
}

// ---------------------------------------------------------------------------
// Kernel 5: LayerNorm (eps=1e-6), one 256-thread block per token.
// ---------------------------------------------------------------------------
__global__ __launch_bounds__(256) void ln_kernel(const float* __restrict__ pre,
                                                 const float* __restrict__ g,
                                                 const float* __restrict__ bta,
                                                 float* __restrict__ y) {
  __shared__ float r1[256];
  __shared__ float r2[256];
  const int t = blockIdx.x, tid = threadIdx.x;
  f4v v = *(const f4v*)&pre[(size_t)t * D_MODEL + tid * 4];
  r1[tid] = v[0] + v[1] + v[2] + v[3];
  r2[tid] = v[0] * v[0] + v[1] * v[1] + v[2] * v[2] + v[3] * v[3];
  __syncthreads();
  for (int off = 128; off > 0; off >>= 1) {
    if (tid < off) {
      r1[tid] += r1[tid + off];
      r2[tid] += r2[tid + off];
    }
    __syncthreads();
  }
  const float mu   = r1[0] * (1.0f / D_MODEL);
  const float var  = r2[0] * (1.0f / D_MODEL) - mu * mu;
  const float rstd = rsqrtf(var + 1e-6f);
  f4v gv = *(const f4v*)&g[tid * 4];
  f4v bv = *(const f4v*)&bta[tid * 4];
  f4v o;
#pragma unroll
  for (int j = 0; j < 4; ++j) o[j] = (v[j] - mu) * rstd * gv[j] + bv[j];
  *(f4v*)&y[(size_t)t * D_MODEL + tid * 4] = o;
}

// ---------------------------------------------------------------------------
extern "C" void kernel_launch(void* const* d_in, const int* in_sizes, int n_in,
                              void* d_out, int out_size, void* d_ws,
                              size_t ws_size, hipStream_t stream) {
  (void)in_sizes; (void)n_in; (void)out_size; (void)ws_size;
  const float* x             = (const float*)d_in[0];
  const unsigned char* mask  = (const unsigned char*)d_in[1];  // jnp bool = u8
  const float* wq            = (const float*)d_in[2];
  const float* wk            = (const float*)d_in[3];
  const float* wv            = (const float*)d_in[4];
  const float* wfc           = (const float*)d_in[5];
  const float* lng           = (const float*)d_in[6];
  const float* lnb           = (const float*)d_in[7];

  float* y    = (float*)d_out;                              // [B,S,D]
  float* attn = (float*)d_out + (size_t)MTOK * D_MODEL;     // [B,H,S,S]

  // workspace: Q,K,V each MTOK*1024 floats (16 MB); ctx/pre alias dead buffers
  float* Qb  = (float*)d_ws;
  float* Kb  = Qb + (size_t)MTOK * D_MODEL;
  float* Vb  = Kb + (size_t)MTOK * D_MODEL;
  float* ctx = Qb;  // Q dead after attn_softmax
  float* pre = Kb;  // K dead after attn_softmax

  qkv_proj_kernel<<<dim3(32, 32, 3), 256, 0, stream>>>(x, wq, wk, wv, Qb, Kb, Vb);
  attn_softmax_kernel<<<dim3(SEQ / 16, NHEAD, 2), 256, 0, stream>>>(Qb, Kb, mask,
                                                                    attn);
  attn_v_kernel<<<dim3(SEQ / 128, NHEAD, 2), 256, 0, stream>>>(attn, Vb, ctx);
  fc_proj_kernel<<<dim3(32, 32, 1), 256, 0, stream>>>(ctx, wfc, x, pre);
  ln_kernel<<<MTOK, 256, 0, stream>>>(pre, lng, lnb, y);
}